// ATA_15908558865236
// MI455X (gfx1250) — compile-verified
//
#include <hip/hip_runtime.h>

typedef __attribute__((ext_vector_type(2))) float v2f;
typedef __attribute__((ext_vector_type(8))) float v8f;

#define C_CH 128
#define LSEQ 2048
#define DK   32
#define BN_EPS 1e-5f

// CDNA5 f32 WMMA: D(16x16) = A(16x4) * B(4x16) + C, wave32.
// A layout: lanes 0-15 row M=lane, K={kk0,kk0+1}; lanes 16-31 row M=lane-16, K={kk0+2,kk0+3}
// B layout: lanes 0-15 col N=lane, K={kk0,kk0+1}; lanes 16-31 col N=lane-16, K={kk0+2,kk0+3}
// D layout: VGPR j = D[M=j + 8*(lane>=16)][N=lane&15]
__device__ __forceinline__ v8f wmma4(v2f a, v2f b, v8f c) {
  return __builtin_amdgcn_wmma_f32_16x16x4_f32(false, a, false, b, (short)0, c, false, false);
}

// ------------------- Conv1d + BN + ReLU via WMMA -------------------
// One block: 16 output positions (x-tile) x all 128 output channels (8 waves x 16).
// GEMM: M = out-channel, N = position, K = in_channel*F + tap (matches OIH weight layout).
// Two accumulators break the WMMA D->C serial chain (2 independent matrix streams).
template<int F, int PAD>
__global__ void conv_bn_relu_kernel(const float* __restrict__ in,   // (8, C, L)
                                    const float* __restrict__ w,    // (C, C, F)
                                    const float* __restrict__ bias,
                                    const float* __restrict__ bns,
                                    const float* __restrict__ bnt,
                                    const float* __restrict__ bnm,
                                    const float* __restrict__ bnu,
                                    float* __restrict__ out)        // (8, C, L), branch offset pre-applied
{
  constexpr int W    = 16 + 2 * PAD;
  constexpr int KTOT = C_CH * F;   // 384 or 1152, both multiples of 8
  __shared__ float patch[C_CH * W];          // <= 12 KB

  const int x0    = blockIdx.x * 16;
  const int batch = blockIdx.y;
  const int tid   = threadIdx.x;
  const float* src = in + (size_t)batch * C_CH * LSEQ;

  for (int e = tid; e < C_CH * W; e += 256) {
    int ch  = e / W;
    int col = e - ch * W;
    int gx  = x0 - PAD + col;
    patch[e] = (gx >= 0 && gx < LSEQ) ? src[ch * LSEQ + gx] : 0.0f;
  }
  __syncthreads();

  const int wave = tid >> 5;
  const int lane = tid & 31;
  const int lrow = lane & 15;
  const int hi   = lane >> 4;
  const int o0   = wave * 16;

  const float* wrow = w + (size_t)(o0 + lrow) * KTOT;
  v8f acc0 = {}, acc1 = {};
  for (int kk0 = 0; kk0 < KTOT; kk0 += 8) {
    int ka = kk0 + 2 * hi;
    int kb = ka + 4;
    v2f a0; a0.x = wrow[ka]; a0.y = wrow[ka + 1];
    v2f a1; a1.x = wrow[kb]; a1.y = wrow[kb + 1];
    int ia0 = ka / F,       fa0 = ka - ia0 * F;
    int ia1 = (ka + 1) / F, fa1 = (ka + 1) - ia1 * F;
    int ib0 = kb / F,       fb0 = kb - ib0 * F;
    int ib1 = (kb + 1) / F, fb1 = (kb + 1) - ib1 * F;
    v2f b0; b0.x = patch[ia0 * W + lrow + fa0];
            b0.y = patch[ia1 * W + lrow + fa1];
    v2f b1; b1.x = patch[ib0 * W + lrow + fb0];
            b1.y = patch[ib1 * W + lrow + fb1];
    acc0 = wmma4(a0, b0, acc0);
    acc1 = wmma4(a1, b1, acc1);
  }
  v8f acc = acc0 + acc1;

  float* dst = out + (size_t)batch * C_CH * LSEQ + x0 + lrow;
  #pragma unroll
  for (int j = 0; j < 8; ++j) {
    int ch = o0 + j + 8 * hi;
    float scale = bns[ch] * rsqrtf(bnu[ch] + BN_EPS);
    float y = (acc[j] + bias[ch] - bnm[ch]) * scale + bnt[ch];
    dst[ch * LSEQ] = fmaxf(y, 0.0f);
  }
}

// ------------------- top-k via full bitonic sort -------------------
// One block per (bb,hh,d) column: gather 4096 vals from the concat-layout
// conv output, sort descending in LDS, emit top 2048 (== lax.top_k sorted).
__global__ void topk_sort_kernel(const float* __restrict__ cat,  // (16, C, L) == concat([br0, br1])
                                 float* __restrict__ outt)       // (8, 4, 2048, 32)
{
  __shared__ float buf[2 * LSEQ];  // 16 KB
  const int id  = blockIdx.x;      // 1024 columns
  const int bb  = id >> 7;
  const int hh  = (id >> 5) & 3;
  const int d   = id & 31;
  const int tid = threadIdx.x;

  const int slab  = 2 * bb + (hh >= 2 ? 1 : 0);   // concat batch index
  const int cbase = (hh & 1) * 64;
  const float* src = cat + (size_t)slab * C_CH * LSEQ;

  for (int r = tid; r < 4096; r += 256) {
    int v = r * 32 + d;
    int c = cbase + (v >> 11);
    int x = v & (LSEQ - 1);
    buf[r] = src[c * LSEQ + x];
  }
  __syncthreads();

  for (int k = 2; k <= 4096; k <<= 1) {
    for (int j = k >> 1; j > 0; j >>= 1) {
      for (int i = tid; i < 4096; i += 256) {
        int l = i ^ j;
        if (l > i) {
          float a = buf[i], bv = buf[l];
          bool desc = ((i & k) == 0);
          if (desc ? (a < bv) : (a > bv)) { buf[i] = bv; buf[l] = a; }
        }
      }
      __syncthreads();
    }
  }

  float* dst = outt + (size_t)((bb * 4 + hh) * LSEQ) * DK + d;
  for (int r = tid; r < LSEQ; r += 256) dst[r * DK] = buf[r];
}

// ------------------- fused attention -------------------
// One block = 16 query rows of one (b,h). Full 16x2048 score strip lives in
// LDS (128 KB, only possible with CDNA5's 320 KB/WGP). attn is written to
// HBM exactly once; context is computed from LDS immediately after.
__global__ void attn_kernel(const float* __restrict__ Qt,
                            const float* __restrict__ Kt,
                            const float* __restrict__ V,
                            float* __restrict__ attn,
                            float* __restrict__ ctx)
{
  extern __shared__ float smem[];
  float* qtile = smem;                     // 16*32   = 512
  float* sc    = smem + 512;               // 16*2048 = 32768
  float* part  = smem + 512 + 16 * LSEQ;   // 8*2*8*32 = 4096

  const int bh   = blockIdx.x >> 7;
  const int m0   = (blockIdx.x & 127) * 16;
  const int tid  = threadIdx.x;
  const int wave = tid >> 5, lane = tid & 31, lrow = lane & 15, hi = lane >> 4;

  const float* qb = Qt + (size_t)bh * LSEQ * DK;
  const float* kb = Kt + (size_t)bh * LSEQ * DK;
  const float* vb = V  + (size_t)bh * LSEQ * DK;

  for (int e = tid; e < 16 * DK; e += 256)
    qtile[e] = qb[(m0 + (e >> 5)) * DK + (e & 31)];
  __syncthreads();

  // A (Q-tile) fragments are loop-invariant across all 128 K-tiles: hoist.
  v2f aq[8];
  #pragma unroll
  for (int s = 0; s < 8; ++s) {
    int kk = s * 4 + 2 * hi;
    aq[s].x = qtile[lrow * DK + kk];
    aq[s].y = qtile[lrow * DK + kk + 1];
  }

  const float inv_sqrt_dk = 0.17677669529663687f;  // 1/sqrt(32)
  for (int nt = wave; nt < 128; nt += 8) {
    int n0 = nt * 16;
    const float* krow = kb + (n0 + lrow) * DK + 2 * hi;
    // speculative prefetch of the wave's next K-tile; OOB is silently dropped
    __builtin_prefetch(kb + (n0 + 128) * DK, 0, 1);
    v8f acc0 = {}, acc1 = {};
    #pragma unroll
    for (int s = 0; s < 8; s += 2) {
      v2f b0; b0.x = krow[s * 4];     b0.y = krow[s * 4 + 1];
      v2f b1; b1.x = krow[s * 4 + 4]; b1.y = krow[s * 4 + 5];
      acc0 = wmma4(aq[s],     b0, acc0);
      acc1 = wmma4(aq[s + 1], b1, acc1);
    }
    v8f acc = acc0 + acc1;
    #pragma unroll
    for (int j = 0; j < 8; ++j)
      sc[(j + 8 * hi) * LSEQ + n0 + lrow] = acc[j] * inv_sqrt_dk;
  }
  __syncthreads();

  // Softmax: 16 lanes per row, wave32 shuffle reductions.
  {
    const int row = tid >> 4, t16 = tid & 15;
    float* r = sc + row * LSEQ;
    float mx = -3.4e38f;
    for (int c = t16; c < LSEQ; c += 16) mx = fmaxf(mx, r[c]);
    for (int off = 8; off; off >>= 1) mx = fmaxf(mx, __shfl_xor(mx, off, 32));
    float sum = 0.f;
    for (int c = t16; c < LSEQ; c += 16) { float e = __expf(r[c] - mx); r[c] = e; sum += e; }
    for (int off = 8; off; off >>= 1) sum += __shfl_xor(sum, off, 32);
    float rinv = 1.0f / sum;
    float* arow = attn + (size_t)(bh * LSEQ + m0 + row) * LSEQ;
    for (int c = t16; c < LSEQ; c += 16) { float p = r[c] * rinv; r[c] = p; arow[c] = p; }
  }
  __syncthreads();

  // Context: attn(16x2048) @ V(2048x32); K split across 8 waves, LDS reduce.
  // Two accumulators -> two independent WMMA chains per wave.
  const int kbeg = wave * 256;
  #pragma unroll
  for (int nt2 = 0; nt2 < 2; ++nt2) {
    int n0 = nt2 * 16;
    v8f acc0 = {}, acc1 = {};
    for (int kk0 = kbeg; kk0 < kbeg + 256; kk0 += 8) {
      int ka = kk0 + 2 * hi;
      int kc = ka + 4;
      v2f a0; a0.x = sc[lrow * LSEQ + ka]; a0.y = sc[lrow * LSEQ + ka + 1];
      v2f a1; a1.x = sc[lrow * LSEQ + kc]; a1.y = sc[lrow * LSEQ + kc + 1];
      v2f b0; b0.x = vb[ka * DK + n0 + lrow]; b0.y = vb[(ka + 1) * DK + n0 + lrow];
      v2f b1; b1.x = vb[kc * DK + n0 + lrow]; b1.y = vb[(kc + 1) * DK + n0 + lrow];
      acc0 = wmma4(a0, b0, acc0);
      acc1 = wmma4(a1, b1, acc1);
    }
    v8f acc = acc0 + acc1;
    #pragma unroll
    for (int j = 0; j < 8; ++j)
      part[((wave * 2 + nt2) * 8 + j) * 32 + lane] = acc[j];
  }
  __syncthreads();

  for (int e = tid; e < 512; e += 256) {
    int row = e >> 5, col = e & 31;
    int nt2 = col >> 4, n = col & 15;
    int j = row & 7, ln = n + ((row >> 3) << 4);
    float s = 0.f;
    #pragma unroll
    for (int wv = 0; wv < 8; ++wv) s += part[((wv * 2 + nt2) * 8 + j) * 32 + ln];
    ctx[(size_t)(bh * LSEQ + m0 + row) * DK + col] = s;
  }
}

extern "C" void kernel_launch(void* const* d_in, const int* in_sizes, int n_in,
                              void* d_out, int out_size, void* d_ws, size_t ws_size,
                              hipStream_t stream) {
  (void)in_sizes; (void)n_in; (void)out_size; (void)ws_size;
  const float* Q = (const float*)d_in[0];
  const float* K = (const float*)d_in[1];
  const float* V = (const float*)d_in[2];
  // per-branch params: w,b,s,t,m,u
  const float* wq0 = (const float*)d_in[3];  const float* pq0[5] = {(const float*)d_in[4],(const float*)d_in[5],(const float*)d_in[6],(const float*)d_in[7],(const float*)d_in[8]};
  const float* wq1 = (const float*)d_in[9];  const float* pq1[5] = {(const float*)d_in[10],(const float*)d_in[11],(const float*)d_in[12],(const float*)d_in[13],(const float*)d_in[14]};
  const float* wk0 = (const float*)d_in[15]; const float* pk0[5] = {(const float*)d_in[16],(const float*)d_in[17],(const float*)d_in[18],(const float*)d_in[19],(const float*)d_in[20]};
  const float* wk1 = (const float*)d_in[21]; const float* pk1[5] = {(const float*)d_in[22],(const float*)d_in[23],(const float*)d_in[24],(const float*)d_in[25],(const float*)d_in[26]};

  // workspace: qcat(16,C,L) | kcat(16,C,L) | Qt(b,h,l,dk) | Kt(b,h,l,dk)  == 48 MB
  float* ws   = (float*)d_ws;
  float* qcat = ws;
  float* kcat = ws + 4194304;
  float* qt   = ws + 8388608;
  float* kt   = ws + 10485760;
  const size_t half = 2097152;  // 8 batches * C * L

  dim3 cgrid(128, 8);
  conv_bn_relu_kernel<3,1><<<cgrid, 256, 0, stream>>>(Q, wq0, pq0[0], pq0[1], pq0[2], pq0[3], pq0[4], qcat);
  conv_bn_relu_kernel<9,4><<<cgrid, 256, 0, stream>>>(Q, wq1, pq1[0], pq1[1], pq1[2], pq1[3], pq1[4], qcat + half);
  conv_bn_relu_kernel<3,1><<<cgrid, 256, 0, stream>>>(K, wk0, pk0[0], pk0[1], pk0[2], pk0[3], pk0[4], kcat);
  conv_bn_relu_kernel<9,4><<<cgrid, 256, 0, stream>>>(K, wk1, pk1[0], pk1[1], pk1[2], pk1[3], pk1[4], kcat + half);

  topk_sort_kernel<<<1024, 256, 0, stream>>>(qcat, qt);
  topk_sort_kernel<<<1024, 256, 0, stream>>>(kcat, kt);

  float* ctx  = (float*)d_out;            // (8,4,2048,32)
  float* attn = (float*)d_out + 2097152;  // (8,4,2048,2048)
  size_t smem = (size_t)(512 + 16 * LSEQ + 4096) * sizeof(float);  // ~146 KB
  attn_kernel<<<4096, 256, smem, stream>>>(qt, kt, V, attn, ctx);
}